// YOLOLossOptimized_6193342841019
// MI455X (gfx1250) — compile-verified
//
#include <hip/hip_runtime.h>
#include <hip/hip_bf16.h>
#include <math.h>

// Problem constants (from reference setup_inputs)
#define B_   8
#define G_   40
#define A_   65536
#define C_   20
#define CH_  25
#define W_   256

typedef float v2f __attribute__((ext_vector_type(2)));
typedef float v8f __attribute__((ext_vector_type(8)));

__device__ __forceinline__ float sigmoidf_(float x){ return 1.f/(1.f + __expf(-x)); }

__device__ __forceinline__ float focal_bce_(float x, float t){
  float p  = sigmoidf_(x);
  float ce = fmaxf(x, 0.f) - x*t + log1pf(__expf(-fabsf(x)));
  float pt = p*t + (1.f-p)*(1.f-t);
  float at = 0.25f*t + 0.75f*(1.f-t);
  float om = 1.f - pt;
  return at*om*om*ce;
}

// ---------------------------------------------------------------------------
// Pass 1: per-anchor decode + class stats; WMMA one-hot projection
//   Delta(A x 20) x OneHot(20 x G)  -> cls-delta pick; fused cost/iou write.
// Block = 256 threads = 8 waves; each wave owns a 16-anchor tile.
// ---------------------------------------------------------------------------
__global__ __launch_bounds__(256) void yolo_pass1(
    const float* __restrict__ in, const float* __restrict__ labels,
    float* __restrict__ cost, float* __restrict__ iouM, float* __restrict__ fgeo){
  const int b    = blockIdx.y;
  const int tid  = threadIdx.x;
  const int wave = tid >> 5;
  const int lane = tid & 31;
  const int half = lane >> 4;      // wave32: lanes 0-15 / 16-31
  const int m    = lane & 15;      // anchor-in-tile (also B-matrix column)
  const int base = blockIdx.x*128 + wave*16;
  const int a    = base + m;

  __shared__ float gts[G_][5];
  __shared__ int   gci[G_];
  __shared__ float sDelta[8][16][21];   // [wave][anchor m][class] (stride 21)
  __shared__ float sD[8][16][16];       // [wave][gt-local n][anchor m]

  for(int i = tid; i < G_*5; i += 256) gts[i/5][i%5] = labels[b*G_*5 + i];
  __syncthreads();
  if(tid < G_) gci[tid] = (int)gts[tid][4];

  const float* inb = in + (size_t)b*CH_*A_;
  float ox = inb[0*A_ + a], oy = inb[1*A_ + a];
  float ow = inb[2*A_ + a], oh = inb[3*A_ + a];
  float oobj = inb[4*A_ + a];
  float xc = (float)(a & (W_-1)) + 0.5f;     // grid x (STRIDE=1)
  float yc = (float)(a >> 8)     + 0.5f;     // grid y
  float px = ox + xc, py = oy + yc;
  float pw = __expf(ow), ph = __expf(oh);
  float sobj = sigmoidf_(oobj);

  // split the 20 classes across the two half-waves
  float snp = 0.f;
  for(int c = half*10; c < half*10 + 10; ++c){
    float s  = sigmoidf_(inb[(5+c)*A_ + a]);
    float p  = sqrtf(s*sobj);
    float lp = __logf(p + 1e-8f);
    float l1 = __logf(1.f - p + 1e-8f);
    sDelta[wave][m][c] = l1 - lp;          // delta = log(1-p) - log(p)
    snp -= l1;                             // sum_neg partial
  }
  float sum_neg = snp + __shfl_xor(snp, 16, 32);

  // fg_geo[a] = OR over all gts of (in_box | in_ctr)
  float fg = 0.f;
  for(int g = 0; g < G_; ++g){
    float dx = fabsf(xc - gts[g][0]);
    float dy = fabsf(yc - gts[g][1]);
    bool ib = (dx < gts[g][2]*0.5f) & (dy < gts[g][3]*0.5f);
    bool ic = (dx < 5.f) & (dy < 5.f);     // CENTER_RADIUS*stride = 5
    if(ib | ic) fg = 1.f;
  }
  __syncthreads();  // sDelta + gci visible to whole block

  float ap  = pw*ph;
  float plx = px - pw*0.5f, ply = py - ph*0.5f;
  float prx = px + pw*0.5f, pry = py + ph*0.5f;

  // 3 gt-tiles of 16 (G=40 padded to 48)
  for(int t = 0; t < 3; ++t){
    const int n0   = t*16;
    const int gcol = n0 + m;
    const int ci   = (gcol < G_) ? gci[gcol] : -1;

    v8f acc = {0.f,0.f,0.f,0.f,0.f,0.f,0.f,0.f};
    #pragma unroll
    for(int k0 = 0; k0 < C_; k0 += 4){
      // A (16x4 f32): lanes<16 -> K=k0,k0+1 ; lanes>=16 -> K=k0+2,k0+3
      v2f Am, Bm;
      Am.x = sDelta[wave][m][k0 + 2*half + 0];
      Am.y = sDelta[wave][m][k0 + 2*half + 1];
      // B (4x16 one-hot of gt class): same half-wave K split, column = m
      Bm.x = (ci == k0 + 2*half + 0) ? 1.f : 0.f;
      Bm.y = (ci == k0 + 2*half + 1) ? 1.f : 0.f;
      acc = __builtin_amdgcn_wmma_f32_16x16x4_f32(
                false, Am, false, Bm, (short)0, acc, false, false);
    }
    __syncthreads();                 // previous tile's sD fully consumed
    #pragma unroll
    for(int r = 0; r < 8; ++r)       // D row = r+8*half, col = m
      sD[wave][m][r + 8*half] = acc[r];
    __syncthreads();

    if(half == 0){                   // lanes 0-15 each own their anchor
      for(int n = 0; n < 16; ++n){
        int g = n0 + n;
        if(g >= G_) break;
        float gx = gts[g][0], gy = gts[g][1], gw = gts[g][2], gh = gts[g][3];
        float tlx = fmaxf(gx - gw*0.5f, plx), tly = fmaxf(gy - gh*0.5f, ply);
        float brx = fminf(gx + gw*0.5f, prx), bry = fminf(gy + gh*0.5f, pry);
        float iw = fmaxf(brx - tlx, 0.f), ih = fmaxf(bry - tly, 0.f);
        float inter = iw*ih;
        float iou = inter / (gw*gh + ap - inter + 1e-16f);
        float iou_cost = -__logf(iou + 1e-8f);
        float dx = fabsf(xc - gx), dy = fabsf(yc - gy);
        bool ib = (dx < gw*0.5f) & (dy < gh*0.5f);
        bool ic = (dx < 5.f) & (dy < 5.f);
        bool in_both = ib & ic;
        float cv = sum_neg + sD[wave][n][m] + 3.f*iou_cost
                 + (in_both ? 0.f : 100000.f)
                 + (fg > 0.f ? 0.f : 1000000.f);
        size_t idx = ((size_t)(b*G_ + g))*A_ + a;
        cost[idx] = cv;
        iouM[idx] = iou;
      }
    }
  }
  if(half == 0) fgeo[(size_t)b*A_ + a] = fg;
}

// ---------------------------------------------------------------------------
// Pass 2: per (b,g) row — top-10 IoU sum -> dyn_k; k-th-smallest cost -> tau
// ---------------------------------------------------------------------------
__device__ __forceinline__ void ins_max10(float* arr, float v){
  if(v <= arr[9]) return;
  int i = 9;
  while(i > 0 && v > arr[i-1]){ arr[i] = arr[i-1]; --i; }
  arr[i] = v;
}
__device__ __forceinline__ void ins_min10(float* arr, float v){
  if(v >= arr[9]) return;
  int i = 9;
  while(i > 0 && v < arr[i-1]){ arr[i] = arr[i-1]; --i; }
  arr[i] = v;
}

__global__ __launch_bounds__(256) void yolo_pass2(
    const float* __restrict__ cost, const float* __restrict__ iouM,
    const float* __restrict__ fgeo, float* __restrict__ tau){
  const int row = blockIdx.x;            // b*G + g
  const int b   = row / G_;
  const int tid = threadIdx.x;
  const float* crow = cost + (size_t)row*A_;
  const float* irow = iouM + (size_t)row*A_;
  const float* frow = fgeo + (size_t)b*A_;

  float tI[10], tC[10];
  for(int i = 0; i < 10; ++i){ tI[i] = 0.f; tC[i] = 3.4e38f; }
  for(int a = tid; a < A_; a += 256){
    float iv = (frow[a] > 0.f) ? irow[a] : 0.f;   // ious_cand
    ins_max10(tI, iv);
    ins_min10(tC, crow[a]);
  }
  __shared__ float sI[256][10];
  __shared__ float sC[256][10];
  for(int i = 0; i < 10; ++i){ sI[tid][i] = tI[i]; sC[tid][i] = tC[i]; }
  __syncthreads();

  __shared__ float mI[8][10];
  __shared__ float mC[8][10];
  if(tid < 8){
    float uI[10], uC[10];
    for(int i = 0; i < 10; ++i){ uI[i] = 0.f; uC[i] = 3.4e38f; }
    for(int j = 0; j < 32; ++j){
      int s = tid*32 + j;
      for(int i = 0; i < 10; ++i){ ins_max10(uI, sI[s][i]); ins_min10(uC, sC[s][i]); }
    }
    for(int i = 0; i < 10; ++i){ mI[tid][i] = uI[i]; mC[tid][i] = uC[i]; }
  }
  __syncthreads();
  if(tid == 0){
    float uI[10], uC[10];
    for(int i = 0; i < 10; ++i){ uI[i] = 0.f; uC[i] = 3.4e38f; }
    for(int j = 0; j < 8; ++j)
      for(int i = 0; i < 10; ++i){ ins_max10(uI, mI[j][i]); ins_min10(uC, mC[j][i]); }
    float s = 0.f;
    for(int i = 0; i < 10; ++i) s += uI[i];
    int k = (int)s;                      // trunc == astype(int32)
    if(k < 1)  k = 1;
    if(k > 10) k = 10;                   // top-10 sum <= 10
    tau[row] = uC[k-1];                  // rank < dyn_k  <=>  cost <= tau
  }
}

// ---------------------------------------------------------------------------
// Pass 3: per-anchor matching + losses; block reduce + global atomics
// accum = { num_fg, sum_iou_loss, sum_obj_loss, sum_cls_loss }
// ---------------------------------------------------------------------------
__global__ __launch_bounds__(256) void yolo_pass3(
    const float* __restrict__ in, const float* __restrict__ labels,
    const float* __restrict__ cost, const float* __restrict__ iouM,
    const float* __restrict__ fgeo, const float* __restrict__ tau,
    float* __restrict__ accum){
  const int b   = blockIdx.y;
  const int tid = threadIdx.x;
  const int a   = blockIdx.x*256 + tid;

  __shared__ float gts[G_][5];
  for(int i = tid; i < G_*5; i += 256) gts[i/5][i%5] = labels[b*G_*5 + i];
  __syncthreads();

  const float* cb = cost + (size_t)b*G_*A_;
  unsigned long long mask = 0ull;
  int best = 0; float bestc = 3.4e38f; int cnt = 0;
  for(int g = 0; g < G_; ++g){
    float cv = cb[(size_t)g*A_ + a];
    if(cv <= tau[b*G_ + g]){ mask |= (1ull << g); ++cnt; }
    if(cv < bestc){ bestc = cv; best = g; }
  }
  if(cnt > 1) mask &= (1ull << best);            // resolve multi-match
  float fgg = fgeo[(size_t)b*A_ + a];
  bool  fg  = (mask != 0ull) && (fgg > 0.f);
  int   mg  = mask ? (__ffsll(mask) - 1) : 0;    // argmax(matching) semantics
  float fgf = fg ? 1.f : 0.f;
  float pred_iou = iouM[((size_t)(b*G_ + mg))*A_ + a];

  const float* inb = in + (size_t)b*CH_*A_;
  float ox = inb[0*A_+a], oy = inb[1*A_+a], ow = inb[2*A_+a], oh = inb[3*A_+a];
  float oobj = inb[4*A_+a];
  float xc = (float)(a & (W_-1)) + 0.5f;
  float yc = (float)(a >> 8)     + 0.5f;
  float px = ox + xc, py = oy + yc, pw = __expf(ow), ph = __expf(oh);

  // elem_iou_loss vs reg_target = gt_boxes[matched_gt]
  float gx = gts[mg][0], gy = gts[mg][1], gw = gts[mg][2], gh = gts[mg][3];
  float tlx = fmaxf(px - pw*0.5f, gx - gw*0.5f);
  float tly = fmaxf(py - ph*0.5f, gy - gh*0.5f);
  float brx = fminf(px + pw*0.5f, gx + gw*0.5f);
  float bry = fminf(py + ph*0.5f, gy + gh*0.5f);
  float en  = ((tlx < brx) && (tly < bry)) ? 1.f : 0.f;
  float ai  = (brx - tlx)*(bry - tly)*en;
  float iou = ai / (pw*ph + gw*gh - ai + 1e-16f);
  float l_iou = (1.f - iou*iou) * fgf;

  float l_obj = focal_bce_(oobj, fgf);

  int mcls = (int)gts[mg][4];
  float l_cls = 0.f;
  for(int c = 0; c < C_; ++c){
    float t = (c == mcls) ? pred_iou*fgf : 0.f;
    l_cls += focal_bce_(inb[(5+c)*A_ + a], t);
  }
  l_cls *= fgf;

  __shared__ float r0[256], r1[256], r2[256], r3[256];
  r0[tid] = fgf; r1[tid] = l_iou; r2[tid] = l_obj; r3[tid] = l_cls;
  __syncthreads();
  for(int s = 128; s > 0; s >>= 1){
    if(tid < s){
      r0[tid] += r0[tid+s]; r1[tid] += r1[tid+s];
      r2[tid] += r2[tid+s]; r3[tid] += r3[tid+s];
    }
    __syncthreads();
  }
  if(tid == 0){
    atomicAdd(&accum[0], r0[0]);
    atomicAdd(&accum[1], r1[0]);
    atomicAdd(&accum[2], r2[0]);
    atomicAdd(&accum[3], r3[0]);
  }
}

// ---------------------------------------------------------------------------
// Pass 4: finalize scalar loss
// ---------------------------------------------------------------------------
__global__ void yolo_pass4(const float* __restrict__ accum, float* __restrict__ out){
  float nfg = fmaxf(accum[0], 1.f);
  out[0] = (5.f*accum[1] + accum[2] + accum[3]) / nfg;   // REG_WEIGHT=5
}

// ---------------------------------------------------------------------------
// Workspace layout (floats): cost[B*G*A] | iou[B*G*A] | fgeo[B*A] | tau[B*G] | accum[4]
// Total ~170 MB.
// ---------------------------------------------------------------------------
extern "C" void kernel_launch(void* const* d_in, const int* in_sizes, int n_in,
                              void* d_out, int out_size, void* d_ws, size_t ws_size,
                              hipStream_t stream){
  (void)in_sizes; (void)n_in; (void)out_size; (void)ws_size;
  const float* in     = (const float*)d_in[0];
  const float* labels = (const float*)d_in[1];
  float* ws = (float*)d_ws;
  const size_t costN = (size_t)B_*G_*A_;
  float* cost  = ws;
  float* iouM  = cost + costN;
  float* fgeo  = iouM + costN;
  float* tau   = fgeo + (size_t)B_*A_;
  float* accum = tau  + B_*G_;

  hipMemsetAsync(accum, 0, 4*sizeof(float), stream);
  yolo_pass1<<<dim3(A_/128, B_), 256, 0, stream>>>(in, labels, cost, iouM, fgeo);
  yolo_pass2<<<B_*G_, 256, 0, stream>>>(cost, iouM, fgeo, tau);
  yolo_pass3<<<dim3(A_/256, B_), 256, 0, stream>>>(in, labels, cost, iouM, fgeo, tau, accum);
  yolo_pass4<<<1, 1, 0, stream>>>(accum, (float*)d_out);
}